// LSTM_34943853920859
// MI455X (gfx1250) — compile-verified
//
#include <hip/hip_runtime.h>
#include <hip/hip_bf16.h>
#include <stdint.h>

typedef __attribute__((ext_vector_type(2))) float v2f;
typedef __attribute__((ext_vector_type(8))) float v8f;

#define LSEQ 512
#define NB   64      // batch
#define ID   1024    // input dim
#define HD   1024    // hidden dim
#define G4   4096    // 4*H
#define KTOT (ID + HD)
#define KC   32      // K chunk staged in LDS
#define LDA  36      // padded LDS row stride (floats), conflict-free for WMMA reads
#define NWG_STEP 128 // workgroups per step kernel (64x32 blocks over [64 x 4096])

// LDS byte address (addrspace(3) offset) of a __shared__ object
__device__ __forceinline__ uint32_t lds_off_of(const float* p) {
    return (uint32_t)(uintptr_t)(const __attribute__((address_space(3))) float*)p;
}

// CDNA5 async copy: 16B global -> LDS, per lane, tracked by ASYNCcnt
__device__ __forceinline__ void async_copy_b128(uint32_t lds_byte, const float* gptr) {
    uint64_t ga = (uint64_t)(uintptr_t)gptr;
    asm volatile("global_load_async_to_lds_b128 %0, %1, off"
                 :: "v"(lds_byte), "v"(ga) : "memory");
}

// ---------------------------------------------------------------------------
// Init: h=h0, c=c0, b = b_ih + b_hh, zero per-step barrier counters.
// ---------------------------------------------------------------------------
__global__ __launch_bounds__(256) void lstm_init(
    const float* __restrict__ h0, const float* __restrict__ c0,
    const float* __restrict__ b_ih, const float* __restrict__ b_hh,
    float* __restrict__ h, float* __restrict__ c,
    float* __restrict__ b, int* __restrict__ counters)
{
    int i = blockIdx.x * 256 + threadIdx.x;      // grid covers 65536
    if (i < NB * HD) { h[i] = h0[i]; c[i] = c0[i]; }
    if (i < G4)      b[i] = b_ih[i] + b_hh[i];
    if (i < LSEQ)    counters[i] = 0;
}

// ---------------------------------------------------------------------------
// One recurrent step:
//   phase 1: z[64,4096] = [xt | h] @ [W_ih | W_hh]^T + b
//            WMMA f32 16x16x4, double-buffered async global->LDS staging
//   phase 2: grid-wide arrive/wait barrier (atomic + s_sleep)
//   phase 3: gate math, update h,c (disjoint slice per WG -> deterministic)
// ---------------------------------------------------------------------------
__global__ __launch_bounds__(256) void lstm_step(
    const float* __restrict__ x,   const float* __restrict__ Wih,
    const float* __restrict__ Whh, const float* __restrict__ bias,
    float* __restrict__ h, float* __restrict__ c,
    float* __restrict__ z, int* __restrict__ counter, int t)
{
    __shared__ float Af[2][64 * LDA];   // A chunk: 64 rows x KC, double buffered
    __shared__ float Bf[2][32 * LDA];   // B chunk: 32 gate-cols x KC

    const int tid  = threadIdx.x;
    const int g    = blockIdx.x;         // 0..127
    const int nbas = g * 32;             // gate-column base of this block
    const int lane = tid & 31;
    const int wave = tid >> 5;           // 0..7
    const int mt   = (wave & 3) * 16;    // M tile base (batch rows)
    const int nt   = (wave >> 2) * 16;   // N tile base within block
    const int lm   = lane & 15;
    const int koff = (lane >> 4) * 2;    // lanes 16-31 hold K=2,3 (ISA 7.12.2)

    // this thread's staging slots (A: two b128, B: one b128) per buffer
    const int ar0 = (tid + 0)   >> 3, ac0 = ((tid + 0)   & 7) * 4;
    const int ar1 = (tid + 256) >> 3, ac1 = ((tid + 256) & 7) * 4;
    const int br  = tid >> 3,         bc  = (tid & 7) * 4;
    uint32_t a0d[2], a1d[2], bd[2];
    #pragma unroll
    for (int u = 0; u < 2; ++u) {
        a0d[u] = lds_off_of(&Af[u][ar0 * LDA + ac0]);
        a1d[u] = lds_off_of(&Af[u][ar1 * LDA + ac1]);
        bd[u]  = lds_off_of(&Bf[u][br  * LDA + bc ]);
    }

    const float* xt = x + (size_t)t * (NB * ID);

    // dual accumulators (k-parity interleave halves the WMMA RAW chain)
    v8f acc0, acc1;
    {
        float bv = bias[nbas + nt + lm];
        #pragma unroll
        for (int j = 0; j < 8; ++j) { acc0[j] = bv; acc1[j] = 0.f; }
    }

    // issue async staging of chunk `ch` into buffer `u`
    auto stage = [&](int u, int ch) {
        const int kg = ch * KC;
        const float* Asrc; const float* Bsrc; int kl;
        if (kg < ID) { Asrc = xt; Bsrc = Wih; kl = kg; }
        else         { Asrc = h;  Bsrc = Whh; kl = kg - ID; }
        async_copy_b128(a0d[u], Asrc + (size_t)ar0 * 1024 + kl + ac0);
        async_copy_b128(a1d[u], Asrc + (size_t)ar1 * 1024 + kl + ac1);
        async_copy_b128(bd[u],  Bsrc + (size_t)(nbas + br) * 1024 + kl + bc);
    };

    const int NCH = KTOT / KC;   // 64
    stage(0, 0);                 // preload chunk 0

    for (int ch = 0; ch < NCH; ++ch) {
        const int cur = ch & 1;
        if (ch + 1 < NCH) {
            stage(cur ^ 1, ch + 1);
            // in-order async completion: <=3 outstanding => chunk `ch` landed
            asm volatile("s_wait_asynccnt 0x3" ::: "memory");
        } else {
            asm volatile("s_wait_asynccnt 0x0" ::: "memory");
        }
        __syncthreads();

        const float* Ab = Af[cur];
        const float* Bb = Bf[cur];
        #pragma unroll
        for (int kk = 0; kk < KC / 4; ++kk) {
            const int k0 = kk * 4 + koff;
            v2f a, b;
            a.x = Ab[(mt + lm) * LDA + k0];
            a.y = Ab[(mt + lm) * LDA + k0 + 1];
            b.x = Bb[(nt + lm) * LDA + k0];
            b.y = Bb[(nt + lm) * LDA + k0 + 1];
            if (kk & 1)
                acc1 = __builtin_amdgcn_wmma_f32_16x16x4_f32(
                    false, a, false, b, (short)0, acc1, false, false);
            else
                acc0 = __builtin_amdgcn_wmma_f32_16x16x4_f32(
                    false, a, false, b, (short)0, acc0, false, false);
        }
        __syncthreads();   // done reading buf[cur] before ch+2 overwrites it
    }

    // write z block: VGPR j -> rows j (lanes 0-15) / j+8 (lanes 16-31)
    {
        const int rhalf = (lane >> 4) * 8;
        const int gc = nbas + nt + lm;
        #pragma unroll
        for (int j = 0; j < 8; ++j) {
            int gr = mt + rhalf + j;
            z[(size_t)gr * G4 + gc] = acc0[j] + acc1[j];
        }
    }

    // ---- grid-wide barrier: arrive (release) then wait (acquire) ----
    __threadfence();
    __syncthreads();
    if (tid == 0) {
        __hip_atomic_fetch_add(counter, 1, __ATOMIC_ACQ_REL, __HIP_MEMORY_SCOPE_AGENT);
        while (__hip_atomic_load(counter, __ATOMIC_ACQUIRE, __HIP_MEMORY_SCOPE_AGENT) < NWG_STEP)
            __builtin_amdgcn_s_sleep(2);
    }
    __syncthreads();
    __threadfence();

    // ---- gate math: 65536 (m, nh) elems, 512 per WG (disjoint slices) ----
    for (int i = g * 512 + tid; i < g * 512 + 512; i += 256) {
        const int m = i >> 10, nh = i & 1023;
        const size_t zb = (size_t)m * G4 + nh;
        float zi = z[zb];
        float zf = z[zb + 1024];
        float zg = z[zb + 2048];
        float zo = z[zb + 3072];
        float si = 1.f / (1.f + __expf(-zi));
        float sf = 1.f / (1.f + __expf(-zf));
        float so = 1.f / (1.f + __expf(-zo));
        float cn = sf * c[i] + si * tanhf(zg);
        float hn = so * tanhf(cn);
        c[i] = cn;
        h[i] = hn;
    }
}

// ---------------------------------------------------------------------------
// Output: y = h @ W_out^T + b_out  (same WMMA tiling, 32 WGs), then copy h,c
// into the concatenated output tuple (y | h | c).
// ---------------------------------------------------------------------------
__global__ __launch_bounds__(256) void lstm_out(
    const float* __restrict__ h, const float* __restrict__ c,
    const float* __restrict__ Wout, const float* __restrict__ bout,
    float* __restrict__ out)
{
    __shared__ float Af[2][64 * LDA];
    __shared__ float Bf[2][32 * LDA];

    const int tid  = threadIdx.x;
    const int nbas = blockIdx.x * 32;     // 32 WGs over 1024 output cols
    const int lane = tid & 31;
    const int wave = tid >> 5;
    const int mt   = (wave & 3) * 16;
    const int nt   = (wave >> 2) * 16;
    const int lm   = lane & 15;
    const int koff = (lane >> 4) * 2;

    const int ar0 = (tid + 0)   >> 3, ac0 = ((tid + 0)   & 7) * 4;
    const int ar1 = (tid + 256) >> 3, ac1 = ((tid + 256) & 7) * 4;
    const int br  = tid >> 3,         bc  = (tid & 7) * 4;
    uint32_t a0d[2], a1d[2], bd[2];
    #pragma unroll
    for (int u = 0; u < 2; ++u) {
        a0d[u] = lds_off_of(&Af[u][ar0 * LDA + ac0]);
        a1d[u] = lds_off_of(&Af[u][ar1 * LDA + ac1]);
        bd[u]  = lds_off_of(&Bf[u][br  * LDA + bc ]);
    }

    v8f acc0, acc1;
    {
        float bv = bout[nbas + nt + lm];
        #pragma unroll
        for (int j = 0; j < 8; ++j) { acc0[j] = bv; acc1[j] = 0.f; }
    }

    auto stage = [&](int u, int ch) {
        const int kl = ch * KC;
        async_copy_b128(a0d[u], h    + (size_t)ar0 * 1024 + kl + ac0);
        async_copy_b128(a1d[u], h    + (size_t)ar1 * 1024 + kl + ac1);
        async_copy_b128(bd[u],  Wout + (size_t)(nbas + br) * 1024 + kl + bc);
    };

    const int NCH = HD / KC;   // 32
    stage(0, 0);

    for (int ch = 0; ch < NCH; ++ch) {
        const int cur = ch & 1;
        if (ch + 1 < NCH) {
            stage(cur ^ 1, ch + 1);
            asm volatile("s_wait_asynccnt 0x3" ::: "memory");
        } else {
            asm volatile("s_wait_asynccnt 0x0" ::: "memory");
        }
        __syncthreads();

        const float* Ab = Af[cur];
        const float* Bb = Bf[cur];
        #pragma unroll
        for (int kk = 0; kk < KC / 4; ++kk) {
            const int k0 = kk * 4 + koff;
            v2f a, b;
            a.x = Ab[(mt + lm) * LDA + k0];
            a.y = Ab[(mt + lm) * LDA + k0 + 1];
            b.x = Bb[(nt + lm) * LDA + k0];
            b.y = Bb[(nt + lm) * LDA + k0 + 1];
            if (kk & 1)
                acc1 = __builtin_amdgcn_wmma_f32_16x16x4_f32(
                    false, a, false, b, (short)0, acc1, false, false);
            else
                acc0 = __builtin_amdgcn_wmma_f32_16x16x4_f32(
                    false, a, false, b, (short)0, acc0, false, false);
        }
        __syncthreads();
    }
    {
        const int rhalf = (lane >> 4) * 8;
        const int gc = nbas + nt + lm;
        #pragma unroll
        for (int j = 0; j < 8; ++j) {
            int gr = mt + rhalf + j;
            out[(size_t)gr * ID + gc] = acc0[j] + acc1[j];   // y [64 x 1024]
        }
    }
    // copy final h and c into output tuple (regions disjoint from y)
    for (int i = blockIdx.x * 256 + tid; i < NB * HD; i += 32 * 256) {
        out[(size_t)NB * ID + i]           = h[i];
        out[(size_t)NB * ID + NB * HD + i] = c[i];
    }
}

// ---------------------------------------------------------------------------
extern "C" void kernel_launch(void* const* d_in, const int* in_sizes, int n_in,
                              void* d_out, int out_size, void* d_ws, size_t ws_size,
                              hipStream_t stream) {
    const float* x    = (const float*)d_in[0];
    const float* h0   = (const float*)d_in[1];
    const float* c0   = (const float*)d_in[2];
    const float* Wih  = (const float*)d_in[3];
    const float* Whh  = (const float*)d_in[4];
    const float* b_ih = (const float*)d_in[5];
    const float* b_hh = (const float*)d_in[6];
    const float* Wout = (const float*)d_in[7];
    const float* bout = (const float*)d_in[8];

    float* ws = (float*)d_ws;
    float* h  = ws;                       // 65536
    float* c  = ws + 65536;               // 65536
    float* b  = ws + 131072;              // 4096
    float* z  = ws + 135168;              // 262144
    int* counters = (int*)(ws + 135168 + 262144);  // 512 ints

    lstm_init<<<256, 256, 0, stream>>>(h0, c0, b_ih, b_hh, h, c, b, counters);

    for (int t = 0; t < LSEQ; ++t) {
        lstm_step<<<NWG_STEP, 256, 0, stream>>>(x, Wih, Whh, b, h, c, z,
                                                counters + t, t);
    }

    lstm_out<<<32, 256, 0, stream>>>(h, c, Wout, bout, (float*)d_out);
}